// CETLossV2_7009386627591
// MI455X (gfx1250) — compile-verified
//
#include <hip/hip_runtime.h>

#define B_SZ    4096
#define D_DIM   2048
#define C_CLS   10000
#define EPSV    0.1f
#define MARGINV 0.3f
#define TK      32    // K-chunk staged in LDS per buffer
#define LDTA    36    // padded LDS row stride in floats: 144B (16B aligned),
                      // 36*r mod 64 distinct for r=0..15 -> conflict-free WMMA reads

typedef float v2f __attribute__((ext_vector_type(2)));
typedef float v8f __attribute__((ext_vector_type(8)));

// ---------------------------------------------------------------------------
// sq[i] = sum_k features[i][k]^2          (one wave32 per row)
// ---------------------------------------------------------------------------
__global__ __launch_bounds__(256) void sqnorm_kernel(const float* __restrict__ f,
                                                     float* __restrict__ sq) {
    const int wave = threadIdx.x >> 5;
    const int lane = threadIdx.x & 31;
    const int row  = blockIdx.x * 8 + wave;
    const float* fr = f + (size_t)row * D_DIM;
    float s = 0.f;
    for (int k = lane; k < D_DIM; k += 32) { float v = fr[k]; s += v * v; }
    #pragma unroll
    for (int m = 16; m > 0; m >>= 1) s += __shfl_xor(s, m, 32);
    if (lane == 0) sq[row] = s;
}

// ---------------------------------------------------------------------------
// Label-smoothed CE per row, single pass (online softmax).
// loss = lse - (1-eps)*logit[tgt] - (eps/C)*sum(logits)
// ---------------------------------------------------------------------------
__global__ __launch_bounds__(256) void idloss_kernel(const float* __restrict__ logits,
                                                     const int* __restrict__ target,
                                                     float* __restrict__ out) {
    const int row = blockIdx.x;
    const float* lr = logits + (size_t)row * C_CLS;
    float m = -3.4e38f, s = 0.f, tot = 0.f;
    for (int c = threadIdx.x; c < C_CLS; c += 256) {
        float x = lr[c];
        tot += x;
        if (x > m) { s = s * __expf(m - x) + 1.f; m = x; }
        else       { s += __expf(x - m); }
    }
    #pragma unroll
    for (int off = 16; off > 0; off >>= 1) {
        float m2 = __shfl_xor(m,   off, 32);
        float s2 = __shfl_xor(s,   off, 32);
        float t2 = __shfl_xor(tot, off, 32);
        float mn = fmaxf(m, m2);
        s = s * __expf(m - mn) + s2 * __expf(m2 - mn);
        m = mn; tot += t2;
    }
    __shared__ float sm[8], ss[8], st[8];
    const int wave = threadIdx.x >> 5;
    if ((threadIdx.x & 31) == 0) { sm[wave] = m; ss[wave] = s; st[wave] = tot; }
    __syncthreads();
    if (threadIdx.x == 0) {
        float M = sm[0], S = ss[0], T = st[0];
        #pragma unroll
        for (int w = 1; w < 8; ++w) {
            float mn = fmaxf(M, sm[w]);
            S = S * __expf(M - mn) + ss[w] * __expf(sm[w] - mn);
            M = mn; T += st[w];
        }
        float lse = M + logf(S);
        float lt  = lr[target[row]];
        out[row]  = lse - (1.f - EPSV) * lt - (EPSV / (float)C_CLS) * T;
    }
}

// ---------------------------------------------------------------------------
// Async global->LDS copy of one 16-byte chunk (CDNA5, tracked by ASYNCcnt).
// lds_byte = addrspace(3) byte offset; gsrc must be 16B aligned.
// ---------------------------------------------------------------------------
__device__ __forceinline__ void async_b128(unsigned lds_byte, const float* gsrc) {
    asm volatile("global_load_async_to_lds_b128 %0, %1, off"
                 :: "v"(lds_byte), "v"(gsrc) : "memory");
}
__device__ __forceinline__ void wait_async0() {
    asm volatile("s_wait_asynccnt 0" ::: "memory");
}

// ---------------------------------------------------------------------------
// Triplet batch-hard mining fused with the fp32 WMMA Gram matrix.
// Block = 64 rows; sweeps all 4096 columns in 64x64 tiles.
// 8 waves: wave w -> i-tile (w&3), j-tiles {2*(w>>2), 2*(w>>2)+1}.
// K staged in double-buffered LDS via global_load_async_to_lds_b128.
// ---------------------------------------------------------------------------
__global__ __launch_bounds__(256) void triplet_kernel(const float* __restrict__ f,
                                                      const float* __restrict__ sq,
                                                      const int* __restrict__ target,
                                                      float* __restrict__ out) {
    __shared__ __align__(16) float As[2][64 * LDTA];
    __shared__ __align__(16) float Bs[2][64 * LDTA];
    __shared__ float sqi[64], sqj[64];
    __shared__ int   tgti[64], tgtj[64];
    __shared__ unsigned mp[64], mn_[64];

    const int tid  = threadIdx.x;
    const int wave = tid >> 5;
    const int lane = tid & 31;
    const int half = lane >> 4;      // K-half select for A/B operands
    const int ln   = lane & 15;      // M (A) / N (B,D) index
    const int ti   = wave & 3;
    const int tj0  = (wave >> 2) * 2;
    const int i0   = blockIdx.x * 64;

    // addrspace(3) byte offsets of the LDS buffers (low 32 bits of generic ptr)
    const unsigned aoff[2] = { (unsigned)(size_t)(void*)&As[0][0],
                               (unsigned)(size_t)(void*)&As[1][0] };
    const unsigned boff[2] = { (unsigned)(size_t)(void*)&Bs[0][0],
                               (unsigned)(size_t)(void*)&Bs[1][0] };
    // this thread's two 16B chunks per array: 512 chunks = 64 rows x 8
    const int idx0 = tid,       r0 = idx0 >> 3, c0 = idx0 & 7;
    const int idx1 = tid + 256, r1 = idx1 >> 3, c1 = idx1 & 7;
    const unsigned l0 = (unsigned)(r0 * (LDTA * 4) + c0 * 16);
    const unsigned l1 = (unsigned)(r1 * (LDTA * 4) + c1 * 16);

    if (tid < 64) {
        sqi[tid]  = sq[i0 + tid];
        tgti[tid] = target[i0 + tid];
        mp[tid]   = 0u;            // running max over positives (dist >= 0)
        mn_[tid]  = 0x7f800000u;   // +inf, running min over negatives
    }

    float lmp[8], lmn[8];
    #pragma unroll
    for (int v = 0; v < 8; ++v) { lmp[v] = 0.f; lmn[v] = 3.4e38f; }

    const int arow  = ti * 16 + ln;         // A row within i-block (M = lane%16)
    const int brow0 = tj0 * 16 + ln;        // B row within j-block, tile 0
    const int brow1 = brow0 + 16;           // tile 1

    const int NCHUNK = D_DIM / TK;          // 64

    for (int jb = 0; jb < B_SZ / 64; ++jb) {
        const int j0 = jb * 64;
        __syncthreads();                    // all reads of prev buffers/sqj done
        if (tid < 64) { sqj[tid] = sq[j0 + tid]; tgtj[tid] = target[j0 + tid]; }

        // prefetch chunk 0 into buffer 0
        async_b128(aoff[0] + l0, f + (size_t)(i0 + r0) * D_DIM + c0 * 4);
        async_b128(boff[0] + l0, f + (size_t)(j0 + r0) * D_DIM + c0 * 4);
        async_b128(aoff[0] + l1, f + (size_t)(i0 + r1) * D_DIM + c1 * 4);
        async_b128(boff[0] + l1, f + (size_t)(j0 + r1) * D_DIM + c1 * 4);

        v8f acc0 = {}; v8f acc1 = {};

        for (int kci = 0; kci < NCHUNK; ++kci) {
            const int cur = kci & 1;
            wait_async0();                  // this wave's async stores landed
            __syncthreads();                // everyone's landed; prev reads done

            if (kci + 1 < NCHUNK) {         // prefetch next chunk -> other buffer
                const int nxt = cur ^ 1;
                const int kn  = (kci + 1) * TK;
                async_b128(aoff[nxt] + l0, f + (size_t)(i0 + r0) * D_DIM + kn + c0 * 4);
                async_b128(boff[nxt] + l0, f + (size_t)(j0 + r0) * D_DIM + kn + c0 * 4);
                async_b128(aoff[nxt] + l1, f + (size_t)(i0 + r1) * D_DIM + kn + c1 * 4);
                async_b128(boff[nxt] + l1, f + (size_t)(j0 + r1) * D_DIM + kn + c1 * 4);
            }

            const float* Ab = &As[cur][0];
            const float* Bb = &Bs[cur][0];
            #pragma unroll
            for (int kk = 0; kk < TK; kk += 4) {
                const int k = kk + half * 2;          // K = 2*(lane/16) + {0,1}
                v2f a;  a.x  = Ab[arow  * LDTA + k]; a.y  = Ab[arow  * LDTA + k + 1];
                v2f b0; b0.x = Bb[brow0 * LDTA + k]; b0.y = Bb[brow0 * LDTA + k + 1];
                v2f b1; b1.x = Bb[brow1 * LDTA + k]; b1.y = Bb[brow1 * LDTA + k + 1];
                acc0 = __builtin_amdgcn_wmma_f32_16x16x4_f32(
                           false, a, false, b0, (short)0, acc0, false, false);
                acc1 = __builtin_amdgcn_wmma_f32_16x16x4_f32(
                           false, a, false, b1, (short)0, acc1, false, false);
            }
        }

        // batch-hard mining on the two 16x16 fp32 tiles of this wave
        #pragma unroll
        for (int v = 0; v < 8; ++v) {
            const int ri = ti * 16 + v + 8 * half;   // D row: M = v + 8*(lane/16)
            const int tr = tgti[ri];
            const float sr = sqi[ri];
            {
                const int cj = tj0 * 16 + ln;
                float d = sqrtf(fmaxf(sr + sqj[cj] - 2.f * acc0[v], 1e-12f));
                if (tgtj[cj] == tr) lmp[v] = fmaxf(lmp[v], d);
                else                lmn[v] = fminf(lmn[v], d);
            }
            {
                const int cj = (tj0 + 1) * 16 + ln;
                float d = sqrtf(fmaxf(sr + sqj[cj] - 2.f * acc1[v], 1e-12f));
                if (tgtj[cj] == tr) lmp[v] = fmaxf(lmp[v], d);
                else                lmn[v] = fminf(lmn[v], d);
            }
        }
    }

    // reduce across the 16 lanes that share a row (N dimension)
    #pragma unroll
    for (int v = 0; v < 8; ++v) {
        #pragma unroll
        for (int off = 8; off > 0; off >>= 1) {
            lmp[v] = fmaxf(lmp[v], __shfl_xor(lmp[v], off, 32));
            lmn[v] = fminf(lmn[v], __shfl_xor(lmn[v], off, 32));
        }
    }
    __syncthreads();
    if (ln == 0) {
        #pragma unroll
        for (int v = 0; v < 8; ++v) {
            const int ri = ti * 16 + v + 8 * half;
            atomicMax(&mp[ri],  __float_as_uint(lmp[v]));   // bits order == value order (>=0)
            atomicMin(&mn_[ri], __float_as_uint(lmn[v]));
        }
    }
    __syncthreads();
    if (tid < 64) {
        float dap = __uint_as_float(mp[tid]);
        float dan = __uint_as_float(mn_[tid]);
        out[i0 + tid] = fmaxf(dap - dan + MARGINV, 0.f);
    }
}

// ---------------------------------------------------------------------------
// out[0] = mean(id rows) + mean(triplet rows) = sum(both)/B
// ---------------------------------------------------------------------------
__global__ __launch_bounds__(256) void finalize_kernel(const float* __restrict__ idl,
                                                       const float* __restrict__ trip,
                                                       float* __restrict__ out) {
    __shared__ float red[256];
    float s = 0.f;
    for (int i = threadIdx.x; i < B_SZ; i += 256) s += idl[i] + trip[i];
    red[threadIdx.x] = s;
    __syncthreads();
    for (int off = 128; off > 0; off >>= 1) {
        if (threadIdx.x < off) red[threadIdx.x] += red[threadIdx.x + off];
        __syncthreads();
    }
    if (threadIdx.x == 0) out[0] = red[0] / (float)B_SZ;
}

extern "C" void kernel_launch(void* const* d_in, const int* in_sizes, int n_in,
                              void* d_out, int out_size, void* d_ws, size_t ws_size,
                              hipStream_t stream) {
    const float* features = (const float*)d_in[0];   // [4096, 2048] f32
    const float* logits   = (const float*)d_in[1];   // [4096, 10000] f32
    const int*   target   = (const int*)d_in[2];     // [4096] i32
    float* out = (float*)d_out;

    float* sq   = (float*)d_ws;       // 4096
    float* idl  = sq  + B_SZ;         // 4096
    float* trip = idl + B_SZ;         // 4096

    sqnorm_kernel  <<<B_SZ / 8,  256, 0, stream>>>(features, sq);
    idloss_kernel  <<<B_SZ,      256, 0, stream>>>(logits, target, idl);
    triplet_kernel <<<B_SZ / 64, 256, 0, stream>>>(features, sq, target, trip);
    finalize_kernel<<<1,         256, 0, stream>>>(idl, trip, out);
}